// Reliable_Memory_63402307223783
// MI455X (gfx1250) — compile-verified
//
#include <hip/hip_runtime.h>

// ---------------------------------------------------------------------------
// Problem constants (fixed by reference setup_inputs)
// ---------------------------------------------------------------------------
constexpr int Bdim = 16;
constexpr int Tdim = 2048;        // power of two: bt >> 11 gives b
constexpr int Cdim = 200;         // classes (M of the GEMM)
constexpr int Ddim = 2048;        // feature dim (N of the GEMM)
constexpr int BT   = Bdim * Tdim; // 32768 (K of the GEMM)
constexpr int Mpad = 208;         // 13 * 16, padded class rows
constexpr float MOM = 0.001f;     // PROTO_MOMENTUM

typedef __attribute__((ext_vector_type(2))) float v2f;
typedef __attribute__((ext_vector_type(8))) float v8f;

// ---------------------------------------------------------------------------
// Kernel 0: zero the per-class counters (deterministic init inside the graph)
// ---------------------------------------------------------------------------
__global__ void rm_zero_counts(unsigned int* __restrict__ counts) {
    counts[threadIdx.x] = 0u;     // 1 block x 256 threads, counts region = 256 u32
}

// ---------------------------------------------------------------------------
// Kernel 1: per-class snippet counts.  Coalesced: 200 consecutive dwords per
// row; integer atomics -> order-independent, bit-deterministic.
// Grid: BT/32 blocks x 256 threads (thread = class column).
// ---------------------------------------------------------------------------
__global__ __launch_bounds__(256) void rm_count_kernel(
        const int* __restrict__ act, const int* __restrict__ vid,
        unsigned int* __restrict__ counts) {
    const int c = threadIdx.x;
    if (c >= Cdim) return;
    const int base = blockIdx.x * 32;
    unsigned int cnt = 0;
    for (int r = 0; r < 32; ++r) {
        const int bt = base + r;
        const int b  = bt >> 11;
        cnt += (act[(size_t)bt * Cdim + c] != 0 && vid[b * Cdim + c] != 0) ? 1u : 0u;
    }
    if (cnt) atomicAdd(&counts[c], cnt);
}

// ---------------------------------------------------------------------------
// Kernel 2: K-split GEMM  partial[kc] += mask^T(16 x Kchunk) * feats(Kchunk x N)
// using V_WMMA_F32_16X16X4_F32.  One wave: 16(M) x 64(N) -> FOUR 16x16
// accumulators sharing a single A (mask) fragment per K-step.
// Strength-reduced addressing: one base pointer per operand, all strides are
// compile-time immediates (act K-step 3200 B, feats K-step 32768 B -- both
// within the 24-bit IOFFSET), single 64-bit pointer bump per 16 K.
// A/B fragment layout (ISA 7.12.2, 32-bit): reg r holds K = 2*(lane>=16)+r,
// lane%16 = M (A) or N (B).  C/D: reg r holds row r + 8*(lane>=16).
// ---------------------------------------------------------------------------
__global__ __launch_bounds__(256) void rm_gemm_kernel(
        const float* __restrict__ feats, const int* __restrict__ act,
        const int* __restrict__ vid, float* __restrict__ part, int klen) {
    const int lane = threadIdx.x & 31;
    const int wave = threadIdx.x >> 5;
    const int half = lane >> 4;   // 0: K=0,1   1: K=2,3
    const int idx  = lane & 15;   // M (for A) / N (for B) within the tile

    const int mbase = blockIdx.x * 16;
    const int nbase = (blockIdx.y * 8 + wave) * 64;
    const int kbase = blockIdx.z * klen;

    const int  c      = mbase + idx;
    const int  cc     = (c < Cdim) ? c : (Cdim - 1);  // clamp: stays in-bounds
    const bool cvalid = (c < Cdim);                   // zero factor kills pad rows

    const int n0 = nbase + idx;                       // subtiles at +0,+16,+32,+48

    v8f acc0 = {}, acc1 = {}, acc2 = {}, acc3 = {};

    // klen is a multiple of T, so each 2048-row segment has a single batch b:
    // hoist the vid_label factor out of the hot loop.
    for (int bt0 = kbase; bt0 < kbase + klen; bt0 += Tdim) {
        const int   b     = bt0 >> 11;
        const float vlabf = (cvalid && vid[b * Cdim + cc] != 0) ? 1.0f : 0.0f;

        // Per-lane base pointers for this 2048-K segment.
        const int*   ap = act   + (size_t)(bt0 + 2 * half) * Cdim + cc;
        const float* fp = feats + (size_t)(bt0 + 2 * half) * Ddim + n0;

        for (int it = 0; it < Tdim / 16; ++it) {      // 128 iters x 16 K
#pragma unroll
            for (int u = 0; u < 4; ++u) {             // 4 K-steps, const offsets
                // A fragment: binary mask in exact f32 (shared by 4 WMMAs)
                const int a0 = ap[u * 4 * Cdim];
                const int a1 = ap[u * 4 * Cdim + Cdim];
                v2f A;
                A.x = a0 ? vlabf : 0.0f;
                A.y = a1 ? vlabf : 0.0f;
                // B fragments: four N sub-tiles (all offsets are immediates)
                const float* fb = fp + u * 4 * Ddim;
                v2f B0; B0.x = fb[0];  B0.y = fb[Ddim + 0];
                v2f B1; B1.x = fb[16]; B1.y = fb[Ddim + 16];
                v2f B2; B2.x = fb[32]; B2.y = fb[Ddim + 32];
                v2f B3; B3.x = fb[48]; B3.y = fb[Ddim + 48];

                acc0 = __builtin_amdgcn_wmma_f32_16x16x4_f32(
                           false, A, false, B0, (short)0, acc0, false, false);
                acc1 = __builtin_amdgcn_wmma_f32_16x16x4_f32(
                           false, A, false, B1, (short)0, acc1, false, false);
                acc2 = __builtin_amdgcn_wmma_f32_16x16x4_f32(
                           false, A, false, B2, (short)0, acc2, false, false);
                acc3 = __builtin_amdgcn_wmma_f32_16x16x4_f32(
                           false, A, false, B3, (short)0, acc3, false, false);
            }
            ap += 16 * Cdim;                          // one u64 add per operand
            fp += 16 * Ddim;                          // per 16 K of work
        }
    }

    // Store this K-chunk's partial tile (fixed location -> deterministic).
    float* p = part + (size_t)blockIdx.z * Mpad * Ddim;
#pragma unroll
    for (int r = 0; r < 8; ++r) {
        const int row = mbase + r + 8 * half;
        float* pr = p + (size_t)row * Ddim + n0;
        pr[0]  = acc0[r];
        pr[16] = acc1[r];
        pr[32] = acc2[r];
        pr[48] = acc3[r];
    }
}

// ---------------------------------------------------------------------------
// Kernel 3: fixed-order K-chunk reduction + prototype EMA epilogue.
// ---------------------------------------------------------------------------
__global__ __launch_bounds__(256) void rm_epilogue_kernel(
        const float* __restrict__ part, const unsigned int* __restrict__ counts,
        const float* __restrict__ proto, float* __restrict__ out, int kc) {
    const int idx = blockIdx.x * blockDim.x + threadIdx.x;
    if (idx >= Cdim * Ddim) return;
    const int c = idx / Ddim;
    const int d = idx - c * Ddim;

    float s = 0.0f;
    for (int k = 0; k < kc; ++k)
        s += part[((size_t)k * Mpad + c) * Ddim + d];

    const unsigned int cnt = counts[c];
    const float pv   = proto[idx];
    const float mean = s / fmaxf((float)cnt, 1.0f);
    out[idx] = (cnt > 0) ? (1.0f - MOM) * pv + MOM * mean : pv;
}

// ---------------------------------------------------------------------------
// Host launcher
// ---------------------------------------------------------------------------
extern "C" void kernel_launch(void* const* d_in, const int* in_sizes, int n_in,
                              void* d_out, int out_size, void* d_ws, size_t ws_size,
                              hipStream_t stream) {
    const float* feats = (const float*)d_in[0];
    const int*   act   = (const int*)d_in[1];
    const int*   vid   = (const int*)d_in[2];
    const float* proto = (const float*)d_in[3];
    float*       out   = (float*)d_out;

    // Choose the K-split from the available workspace (deterministic given
    // constant ws_size).  Chunks stay multiples of T so the vid factor hoists.
    const size_t per_chunk = (size_t)Mpad * Ddim * sizeof(float);
    int KC = 16;
    while (KC > 1 && (size_t)KC * per_chunk + 256 * sizeof(unsigned int) > ws_size)
        KC >>= 1;

    float*        part   = (float*)d_ws;
    unsigned int* counts = (unsigned int*)((char*)d_ws + (size_t)KC * per_chunk);

    rm_zero_counts<<<dim3(1), dim3(256), 0, stream>>>(counts);
    rm_count_kernel<<<dim3(BT / 32), dim3(256), 0, stream>>>(act, vid, counts);
    rm_gemm_kernel<<<dim3(Mpad / 16, Ddim / 512, KC), dim3(256), 0, stream>>>(
        feats, act, vid, part, BT / KC);
    rm_epilogue_kernel<<<dim3((Cdim * Ddim + 255) / 256), dim3(256), 0, stream>>>(
        part, counts, proto, out, KC);
}